// NDE_loss_21981642621253
// MI455X (gfx1250) — compile-verified
//
#include <hip/hip_runtime.h>
#include <stdint.h>

// ---------------------------------------------------------------------------
// NDE loss: loss = mean_n( -[ mean_e( mean_t(ls_ori) - mean_t(ls_gen) ) ] )
// where ls_* = logits gathered at entity ids minus logsumexp over V.
// Bandwidth-bound: ~618 MB of f32 logits -> ~27us floor @ 23.3 TB/s.
// Phase 1 streams rows through LDS with CDNA5 async copies (ASYNCcnt),
// chunk-max-then-exp => ~1.06 v_exp_f32 per element (stays under mem floor).
// Phase 2 is a tiny gather + reduction.
// ---------------------------------------------------------------------------

#define L2E_F 1.4426950408889634f
#define LN2_F 0.6931471805599453f

constexpr int TPB    = 256;        // 8 waves (wave32)
constexpr int C4     = 1024;       // float4 per chunk = 4096 floats = 16 KB
constexpr int ISSUES = C4 / TPB;   // 4 async b128 issues per thread per chunk

__device__ __forceinline__ float fast_exp2(float x) {
#if __has_builtin(__builtin_amdgcn_exp2f)
  return __builtin_amdgcn_exp2f(x);      // v_exp_f32
#else
  return exp2f(x);
#endif
}
__device__ __forceinline__ float fast_log2(float x) {
#if __has_builtin(__builtin_amdgcn_logf)
  return __builtin_amdgcn_logf(x);       // v_log_f32 (log2)
#else
  return log2f(x);
#endif
}

// CDNA5 async global->LDS copy, 16B per lane, tracked by ASYNCcnt.
__device__ __forceinline__ void async_copy_f4(const float4* gaddr, uint32_t lds_byte) {
  asm volatile("global_load_async_to_lds_b128 %0, %1, off"
               :: "v"(lds_byte), "v"(gaddr)
               : "memory");
}

// One block per row; computes lse[row] = logsumexp_e(src[row, 0..V)).
// Rows [0, oriRows) come from `ori`, the rest from `cap`.
__global__ __launch_bounds__(TPB) void lse_rows_kernel(
    const float* __restrict__ ori, const float* __restrict__ cap,
    float* __restrict__ lse, int oriRows, int V) {
  __shared__ float4 sbuf[2][C4];   // 32 KB double buffer
  __shared__ float  redM[TPB];
  __shared__ float  redS[TPB];

  const int row = blockIdx.x;
  const int tid = threadIdx.x;
  const float* src = (row < oriRows) ? (ori + (size_t)row * V)
                                     : (cap + (size_t)(row - oriRows) * V);
  const float4* src4 = (const float4*)src;
  const int V4  = V >> 2;
  const int NCH = (V4 + C4 - 1) / C4;

  // Issue one chunk: every lane always issues (addresses clamped) so the
  // per-wave ASYNCcnt arithmetic below is exact (4 issues per chunk).
  auto issue = [&](int c) {
    const int sel = c & 1;
    const uint32_t lds0 = (uint32_t)(uintptr_t)(&sbuf[sel][tid]);
#pragma unroll
    for (int j = 0; j < ISSUES; ++j) {
      int f4  = c * C4 + j * TPB + tid;
      int idx = (f4 < V4) ? f4 : (V4 - 1);          // clamp (dupes unread)
      async_copy_f4(src4 + idx, lds0 + (uint32_t)(j * TPB) * 16u);
    }
  };

  float m = -INFINITY, s = 0.0f;   // running base-2 logsumexp state

  issue(0);
  for (int c = 0; c < NCH; ++c) {
    if (c + 1 < NCH) {
      issue(c + 1);                                    // prefetch next chunk
      asm volatile("s_wait_asynccnt 4" ::: "memory");  // chunk c landed
    } else {
      asm volatile("s_wait_asynccnt 0" ::: "memory");
    }
    const int sel  = c & 1;
    const int base = c * C4;

    // Each thread reads back exactly the float4s its own lane copied.
    float4 r[ISSUES];
    int   nv  = 0;               // valid j's form a prefix (f4 grows with j)
    float m16 = -INFINITY;
#pragma unroll
    for (int j = 0; j < ISSUES; ++j) {
      int f4 = base + j * TPB + tid;
      if (f4 < V4) {
        float4 v = sbuf[sel][j * TPB + tid];
        v.x *= L2E_F; v.y *= L2E_F; v.z *= L2E_F; v.w *= L2E_F;
        r[j] = v;
        m16 = fmaxf(m16, fmaxf(fmaxf(v.x, v.y), fmaxf(v.z, v.w)));
        nv = j + 1;
      }
    }
    if (nv) {
      float mnew = fmaxf(m, m16);
      s *= fast_exp2(m - mnew);          // one rescale per chunk per thread
#pragma unroll
      for (int j = 0; j < ISSUES; ++j) {
        if (j < nv) {
          s += fast_exp2(r[j].x - mnew);
          s += fast_exp2(r[j].y - mnew);
          s += fast_exp2(r[j].z - mnew);
          s += fast_exp2(r[j].w - mnew);
        }
      }
      m = mnew;
    }
  }

  // Scalar tail (V % 4 elements); stays inside the allocation.
  if (tid == 0) {
    for (int k = V4 << 2; k < V; ++k) {
      float y = src[k] * L2E_F;
      float mnew = fmaxf(m, y);
      s = s * fast_exp2(m - mnew) + fast_exp2(y - mnew);
      m = mnew;
    }
  }

  // Block-wide (m, s) merge.
  redM[tid] = m; redS[tid] = s;
  __syncthreads();
  for (int off = TPB >> 1; off > 0; off >>= 1) {
    if (tid < off) {
      float m1 = redM[tid], m2 = redM[tid + off];
      float M  = fmaxf(m1, m2);
      redS[tid] = redS[tid] * fast_exp2(m1 - M) + redS[tid + off] * fast_exp2(m2 - M);
      redM[tid] = M;
    }
    __syncthreads();
  }
  if (tid == 0)
    lse[row] = (redM[0] + fast_log2(redS[0])) * LN2_F;   // back to natural log
}

// Tiny gather + final mean. lse layout: [0,B0*T) ori rows, then N*T cap rows.
__global__ __launch_bounds__(256) void finalize_kernel(
    const float* __restrict__ ori, const float* __restrict__ cap,
    const int* __restrict__ lens, const int* __restrict__ ids,
    const float* __restrict__ lse, float* __restrict__ out,
    int B0, int N, int E, int T, int V) {
  const int tid   = threadIdx.x;
  const int total = N * E * T;
  float acc = 0.0f;
  for (int i = tid; i < total; i += blockDim.x) {
    int n = i / (E * T);
    int e = (i / T) % E;
    int t = i % T;
    // repeated[n]: repeat_interleave(arange(B0), lens)
    int b = B0 - 1, run = 0;
    for (int bb = 0; bb < B0; ++bb) { run += lens[bb]; if (n < run) { b = bb; break; } }
    int id = ids[n * E + e];
    float go = ori[((size_t)(b * T + t)) * V + id] - lse[b * T + t];
    float gg = cap[((size_t)(n * T + t)) * V + id] - lse[B0 * T + n * T + t];
    acc += (go - gg);
  }
  __shared__ float red[256];
  red[tid] = acc;
  __syncthreads();
  for (int off = 128; off > 0; off >>= 1) {
    if (tid < off) red[tid] += red[tid + off];
    __syncthreads();
  }
  if (tid == 0) out[0] = -red[0] / (float)total;
}

extern "C" void kernel_launch(void* const* d_in, const int* in_sizes, int n_in,
                              void* d_out, int out_size, void* d_ws, size_t ws_size,
                              hipStream_t stream) {
  (void)n_in; (void)out_size; (void)ws_size;
  const float* ori  = (const float*)d_in[0];   // [B0, T, V] f32
  const float* cap  = (const float*)d_in[1];   // [N,  T, V] f32
  const int*   lens = (const int*)d_in[2];     // [B0]
  const int*   ids  = (const int*)d_in[3];     // [N, E]

  const int V  = 50257;                        // reference vocab size
  const int B0 = in_sizes[2];
  const int N  = in_sizes[4];
  const int E  = in_sizes[3] / N;
  const int T  = in_sizes[0] / (B0 * V);

  float* lse = (float*)d_ws;                   // (B0+N)*T floats
  const int R = (B0 + N) * T;

  lse_rows_kernel<<<R, TPB, 0, stream>>>(ori, cap, lse, B0 * T, V);
  finalize_kernel<<<1, 256, 0, stream>>>(ori, cap, lens, ids, lse,
                                         (float*)d_out, B0, N, E, T, V);
}